// SpatialSO_71433896067634
// MI455X (gfx1250) — compile-verified
//
#include <hip/hip_runtime.h>

typedef __attribute__((ext_vector_type(2))) float v2f;
typedef __attribute__((ext_vector_type(8))) float v8f;

#define BDIM 16
#define CDIM 64
#define HDIM 512
#define WDIM 512
#define N2   256   // 16*16 pooled bins

// ---------------------------------------------------------------------------
// Kernel 1: adaptive avg pool 512x512 -> 16x16 per (b,c) image.
// One workgroup (256 threads = 8 wave32) per image. Fully coalesced float4
// streaming reads; groups of 8 lanes share one 32-column bin and reduce via
// shfl_xor, then a single LDS atomic add per group.
// ---------------------------------------------------------------------------
__global__ __launch_bounds__(256) void pool_kernel(const float* __restrict__ x,
                                                   float* __restrict__ p)
{
    const int img = blockIdx.x;                      // b*CDIM + c
    const float4* __restrict__ src4 =
        (const float4*)(x + (size_t)img * (HDIM * (size_t)WDIM));

    __shared__ float bins[N2];
    const int t = threadIdx.x;
    bins[t] = 0.0f;
    __syncthreads();

    const int rofs = t >> 7;      // which of the two rows this iteration
    const int c4   = t & 127;     // float4 column (128 per row)
    const int jx   = c4 >> 3;     // 8 float4 (=32 floats) per x-bin

    for (int y0 = 0; y0 < HDIM; y0 += 2) {
        const int r = y0 + rofs;
        const float4 v = src4[(size_t)r * (WDIM / 4) + c4];
        float s = v.x + v.y + v.z + v.w;
        // reduce the 8 lanes that share (row, jx)
        s += __shfl_xor(s, 1);
        s += __shfl_xor(s, 2);
        s += __shfl_xor(s, 4);
        if ((t & 7) == 0)
            atomicAdd(&bins[(r >> 5) * 16 + jx], s);
    }
    __syncthreads();
    p[(size_t)img * N2 + t] = bins[t] * (1.0f / (32.0f * 32.0f));
}

// ---------------------------------------------------------------------------
// Kernel 2: per batch, q = P(64x256) * W^T(256x256) via v_wmma_f32_16x16x4_f32,
// fused with out[b,i] = (1/64) * sum_c p[c,i]*q[c,i] + bias[i].
// One workgroup per batch, 8 waves; each wave handles 2 N-tiles.
//
// Fragment layouts (CDNA5 ISA 7.12.2, wave32):
//   A 16x4 f32 : lane L, vgpr v -> M = L&15, K = v + 2*(L>>4)
//   B 4x16 f32 : lane L, vgpr v -> N = L&15, K = v + 2*(L>>4)
//   C/D 16x16  : lane L, vgpr v -> M = v + 8*(L>>4), N = L&15
// ---------------------------------------------------------------------------
__global__ __launch_bounds__(256) void gemm_kernel(const float* __restrict__ p,
                                                   const float* __restrict__ w,
                                                   const float* __restrict__ bias,
                                                   float* __restrict__ outv)
{
    const int b = blockIdx.x;
    const float* __restrict__ pb = p + (size_t)b * (CDIM * N2);

    const int t  = threadIdx.x;
    const int wv = t >> 5;    // wave id 0..7
    const int L  = t & 31;    // lane id
    const int lo = L & 15;
    const int hi = L >> 4;

    for (int n = wv; n < 16; n += 8) {
        const int i = 16 * n + lo;       // output column for this lane (pair)
        float s = 0.0f;

        for (int m = 0; m < 4; ++m) {    // c-tiles: rows 16m..16m+15
            v8f acc = {};
            // A(c,j) = pb[c*256 + j], row c = 16m + lo, K base offset 2*hi
            const float* __restrict__ arow = pb + (size_t)(16 * m + lo) * N2 + 2 * hi;
            // B(j,i) = W^T => w[i*256 + j], row i, K base offset 2*hi
            const float* __restrict__ brow = w + (size_t)i * N2 + 2 * hi;

#pragma unroll 4
            for (int k0 = 0; k0 < 64; ++k0) {
                v2f a  = *(const v2f*)(arow + 4 * k0);
                v2f bf = *(const v2f*)(brow + 4 * k0);
                acc = __builtin_amdgcn_wmma_f32_16x16x4_f32(
                        false, a, false, bf, (short)0, acc, false, false);
            }

            // acc[v] = q[c][i] with c = 16m + v + 8*hi; fuse sum_c p[c][i]*q[c][i]
#pragma unroll
            for (int v = 0; v < 8; ++v) {
                const int c = 16 * m + v + 8 * hi;
                s += acc[v] * pb[(size_t)c * N2 + (16 * n + lo)];
            }
        }
        // combine the two lane-halves (M = 0..7+.. vs 8..15+..)
        s += __shfl_xor(s, 16);
        if (hi == 0)
            outv[(size_t)b * N2 + i] = s * (1.0f / CDIM) + bias[i];
    }
}

// ---------------------------------------------------------------------------
// Kernel 3: nearest upsample out[16,256] -> y[16,1,512,512]. One float4 store
// per thread; every float4 lies inside a single 32-column bin.
// ---------------------------------------------------------------------------
__global__ __launch_bounds__(256) void upsample_kernel(const float* __restrict__ outv,
                                                       float4* __restrict__ y)
{
    const int idx = blockIdx.x * 256 + threadIdx.x;   // global float4 index
    const int b   = idx >> 16;                        // 65536 float4 per image
    const int rem = idx & 65535;
    const int h   = rem >> 7;                         // 128 float4 per row
    const int w4  = rem & 127;
    const int bin = (h >> 5) * 16 + (w4 >> 3);
    const float v = outv[b * N2 + bin];
    y[idx] = make_float4(v, v, v, v);
}

extern "C" void kernel_launch(void* const* d_in, const int* in_sizes, int n_in,
                              void* d_out, int out_size, void* d_ws, size_t ws_size,
                              hipStream_t stream)
{
    const float* x    = (const float*)d_in[0];   // [16,64,512,512]
    const float* w    = (const float*)d_in[1];   // [256,256]
    const float* bias = (const float*)d_in[2];   // [256]
    float* y          = (float*)d_out;           // [16,1,512,512]

    float* p    = (float*)d_ws;                  // [16,64,256] = 1 MiB
    float* outv = p + (size_t)BDIM * CDIM * N2;  // [16,256]

    pool_kernel<<<BDIM * CDIM, 256, 0, stream>>>(x, p);
    gemm_kernel<<<BDIM, 256, 0, stream>>>(p, w, bias, outv);

    const int n4 = BDIM * HDIM * WDIM / 4;       // 1,048,576 float4
    upsample_kernel<<<n4 / 256, 256, 0, stream>>>(outv, (float4*)y);
}